// ThreeBodySpringMassGraphModel_20358144983602
// MI455X (gfx1250) — compile-verified
//
#include <hip/hip_runtime.h>

// ThreeBodySpringMass graph network, fully fused, fp32 WMMA (gfx1250).
// Stage-major schedule: each GEMM's weights are staged transposed into LDS
// once per workgroup, then all tiles run WMMA with ds_load_b64 B-fragments.

#define B_TOT  16384
#define NNODE  3
#define DDIM   2
#define HID    128
#define BCHUNK 16
#define NROWS  (BCHUNK * NNODE)          // 48 node rows / chunk
#define EROWS  (BCHUNK * NNODE * NNODE)  // 144 edge rows / chunk
#define STR    132                       // activation row stride (floats)

// Wt (transposed weights) column strides: must be == 4 (mod 64) for
// conflict-free b64 B-fragment loads, and >= staged K.
#define WS_S   68                        // small K (<=64)
#define WS_128 132
#define WS_256 260
#define WS_384 388

#define WT_FLOATS (HID * WS_384)         // 49664 floats (biggest: Wm1, K=384)
#define NODET_OFF (WT_FLOATS - NROWS * STR) // nodeT aliases tail of WtBig (43328)

typedef float v2f __attribute__((ext_vector_type(2)));
typedef float v8f __attribute__((ext_vector_type(8)));

struct PSet {
  const float *Wn1, *bn1, *Wn2, *bn2;
  const float *We1, *be1, *We2, *be2;
  const float *Wm1, *bm1, *Wm2, *bm2;
  const float *Wu1, *bu1, *Wu2, *bu2;
  const float *Wd,  *bd;
};

__device__ __forceinline__ void zero_acc(v8f* c) {
  v8f z = {0.f, 0.f, 0.f, 0.f, 0.f, 0.f, 0.f, 0.f};
#pragma unroll
  for (int i = 0; i < 8; ++i) c[i] = z;
}

// Stage W[Krows x 128] (row-major, global) into LDS transposed + zero-padded:
//   wt[col*wstr + kk] = (kk < Krows) ? W[kk*128 + col] : 0   for kk < Kpad
// 128 threads: thread == column; global reads coalesced across threads.
__device__ __forceinline__ void stage_w(float* wt, const float* __restrict__ W,
                                        int Krows, int Kpad, int wstr, int tid) {
  float* dst = wt + tid * wstr;
  for (int kk = 0; kk < Kpad; ++kk)
    dst[kk] = (kk < Krows) ? W[kk * HID + tid] : 0.f;
}

// C[16x128] += A[16xKpad] (LDS rows, per-lane row ptr) x Wt (LDS, transposed).
// A frag: lane l = row M=l%16, K pair k0 + 2*(l/16) + {0,1}
// B frag: lane l = col N=nt*16 + l%16, same K pair -> contiguous b64 in Wt.
__device__ __forceinline__ void mm_lds(v8f* c, const float* aRow, int kOff,
                                       const float* wt, int wstr, int nsel, int Kpad) {
#pragma unroll 4
  for (int k0 = 0; k0 < Kpad; k0 += 4) {
    v2f a = *(const v2f*)(aRow + k0 + kOff);
    const float* wb = wt + nsel * wstr + k0 + kOff;
#pragma unroll
    for (int nt = 0; nt < 8; ++nt) {
      v2f b = *(const v2f*)(wb + nt * 16 * wstr);
      c[nt] = __builtin_amdgcn_wmma_f32_16x16x4_f32(
          false, a, false, b, (short)0, c[nt], false, false);
    }
  }
}

// Store C tile (16x128) to LDS rows at dst, adding bias, optional relu.
// C layout: lane l holds N = 16*nt + l%16, M = 8*(l/16) + v.
__device__ __forceinline__ void store_tile(float* dst, int lane, const v8f* c,
                                           const float* __restrict__ bias, bool relu) {
  const int nsel = lane & 15, grp = lane >> 4;
#pragma unroll
  for (int nt = 0; nt < 8; ++nt) {
    float bv = bias[nt * 16 + nsel];
#pragma unroll
    for (int v = 0; v < 8; ++v) {
      float val = c[nt][v] + bv;
      if (relu) val = fmaxf(val, 0.f);
      dst[(grp * 8 + v) * STR + nt * 16 + nsel] = val;
    }
  }
}

__global__ __launch_bounds__(128) void gnet_fused_kernel(
    const float* __restrict__ dq1, const float* __restrict__ dq2,
    const float* __restrict__ dp1, const float* __restrict__ dp2,
    const float* __restrict__ mArr, const float* __restrict__ lenArr,
    const float* __restrict__ kArr, PSet pq, PSet pp,
    float* __restrict__ out) {
  extern __shared__ float smem[];
  float* hnB   = smem;                    // NROWS*STR : encoded nodes hn
  float* msgB  = hnB + NROWS * STR;       // EROWS*STR : edge-attr/he/msg/hu1
  float* wt    = msgB + EROWS * STR;      // WT_FLOATS : staged weights
  float* nodeT = wt + NODET_OFF;          // NROWS*STR : node temp / agg (aliases Wt tail)

  const int tid  = threadIdx.x;
  const int lane = tid & 31;
  const int wave = tid >> 5;
  const int nsel = lane & 15;
  const int kOff = (lane >> 4) * 2;

  const PSet P  = (blockIdx.y == 0) ? pq : pp;
  const int nr0 = blockIdx.x * NROWS;
  const int er0 = blockIdx.x * EROWS;

  // ---- fill node/edge attrs (zero-padded) + stage Wn1 ----
  if (tid < NROWS) {
    const int ng = nr0 + tid;
    float* r = nodeT + tid * STR;
    r[0] = dq1[ng * 2 + 0]; r[1] = dq2[ng * 2 + 0];
    r[2] = dp1[ng * 2 + 0]; r[3] = dp2[ng * 2 + 0];
    r[4] = dq1[ng * 2 + 1]; r[5] = dq2[ng * 2 + 1];
    r[6] = dp1[ng * 2 + 1]; r[7] = dp2[ng * 2 + 1];
    r[8] = mArr[ng]; r[9] = 0.f; r[10] = 0.f; r[11] = 0.f;
  }
  for (int r = tid; r < EROWS; r += 128) {
    const int eg = er0 + r;
    float* e = msgB + r * STR;
    e[0] = lenArr[eg]; e[1] = kArr[eg]; e[2] = 0.f; e[3] = 0.f;
  }
  stage_w(wt, P.Wn1, 9, 12, WS_S, tid);
  __syncthreads();

  // ---- hn layer1 (K=12, relu) in-place nodeT ----
  if (wave < 3) {
    v8f c[8]; zero_acc(c);
    mm_lds(c, nodeT + (wave * 16 + nsel) * STR, kOff, wt, WS_S, nsel, 12);
    store_tile(nodeT + wave * 16 * STR, lane, c, P.bn1, true);
  }
  __syncthreads();
  stage_w(wt, P.Wn2, HID, HID, WS_128, tid);
  __syncthreads();

  // ---- hn layer2 -> hnB ----
  if (wave < 3) {
    v8f c[8]; zero_acc(c);
    mm_lds(c, nodeT + (wave * 16 + nsel) * STR, kOff, wt, WS_128, nsel, HID);
    store_tile(hnB + wave * 16 * STR, lane, c, P.bn2, false);
  }
  __syncthreads();
  stage_w(wt, P.We1, 2, 4, WS_S, tid);
  __syncthreads();

  // ---- he layer1 (K=4, relu) in-place msgB ----
  for (int t = wave; t < 9; t += 4) {
    v8f c[8]; zero_acc(c);
    mm_lds(c, msgB + (t * 16 + nsel) * STR, kOff, wt, WS_S, nsel, 4);
    store_tile(msgB + t * 16 * STR, lane, c, P.be1, true);
  }
  __syncthreads();
  stage_w(wt, P.We2, HID, HID, WS_128, tid);
  __syncthreads();

  // ---- he layer2 in-place msgB ----
  for (int t = wave; t < 9; t += 4) {
    v8f c[8]; zero_acc(c);
    mm_lds(c, msgB + (t * 16 + nsel) * STR, kOff, wt, WS_128, nsel, HID);
    store_tile(msgB + t * 16 * STR, lane, c, P.be2, false);
  }
  __syncthreads();
  stage_w(wt, P.Wm1, 3 * HID, 3 * HID, WS_384, tid);   // full K=384 resident
  __syncthreads();

  // ---- msg layer1: K=384 = [he | hn[sender] | hn[receiver]], relu, in-place ----
  for (int t = wave; t < 9; t += 4) {
    const int el = t * 16 + nsel;          // this lane's A-row (local edge index)
    const int bl = el / 9, q = el % 9;
    const int srow = bl * 3 + (q % 3);     // sender   = e % 3
    const int rrow = bl * 3 + (q / 3);     // receiver = e / 3
    v8f c[8]; zero_acc(c);
    mm_lds(c, msgB + el * STR,   kOff, wt,           WS_384, nsel, HID);
    mm_lds(c, hnB + srow * STR,  kOff, wt + HID,     WS_384, nsel, HID);
    mm_lds(c, hnB + rrow * STR,  kOff, wt + 2 * HID, WS_384, nsel, HID);
    store_tile(msgB + t * 16 * STR, lane, c, P.bm1, true);
  }
  __syncthreads();
  stage_w(wt, P.Wm2, HID, HID, WS_128, tid);
  __syncthreads();

  // ---- msg layer2 in-place msgB ----
  for (int t = wave; t < 9; t += 4) {
    v8f c[8]; zero_acc(c);
    mm_lds(c, msgB + (t * 16 + nsel) * STR, kOff, wt, WS_128, nsel, HID);
    store_tile(msgB + t * 16 * STR, lane, c, P.bm2, false);
  }
  __syncthreads();

  // ---- aggregate (3 consecutive msg rows per node) -> nodeT ; stage Wu1 ----
  for (int idx = tid; idx < NROWS * HID; idx += 128) {
    const int row = idx >> 7, col = idx & 127;
    nodeT[row * STR + col] = msgB[(row * 3 + 0) * STR + col] +
                             msgB[(row * 3 + 1) * STR + col] +
                             msgB[(row * 3 + 2) * STR + col];
  }
  stage_w(wt, P.Wu1, 2 * HID, 2 * HID, WS_256, tid);   // full K=256 resident
  __syncthreads();

  // ---- hu layer1: K=256 = [hn | agg], relu -> msgB rows 0..47 ----
  if (wave < 3) {
    v8f c[8]; zero_acc(c);
    mm_lds(c, hnB   + (wave * 16 + nsel) * STR, kOff, wt,       WS_256, nsel, HID);
    mm_lds(c, nodeT + (wave * 16 + nsel) * STR, kOff, wt + HID, WS_256, nsel, HID);
    store_tile(msgB + wave * 16 * STR, lane, c, P.bu1, true);
  }
  __syncthreads();
  stage_w(wt, P.Wu2, HID, HID, WS_128, tid);
  __syncthreads();

  // ---- hu layer2 -> nodeT, then decode out = hu @ Wd[128,2] + bd ----
  if (wave < 3) {
    v8f c[8]; zero_acc(c);
    mm_lds(c, msgB + (wave * 16 + nsel) * STR, kOff, wt, WS_128, nsel, HID);
    store_tile(nodeT + wave * 16 * STR, lane, c, P.bu2, false);

    // 16 rows x 2 outputs == 32 lanes; per-lane dot over 128 (float4 LDS reads)
    const int rowl = lane >> 1, dsel = lane & 1;
    const float4* hr = (const float4*)(nodeT + (wave * 16 + rowl) * STR);
    float acc = P.bd[dsel];
#pragma unroll 8
    for (int kk = 0; kk < 32; ++kk) {
      float4 h = hr[kk];
      const float* wd = P.Wd + kk * 8 + dsel;
      acc += h.x * wd[0] + h.y * wd[2] + h.z * wd[4] + h.w * wd[6];
    }
    const int ng = nr0 + wave * 16 + rowl;
    out[(size_t)blockIdx.y * (B_TOT * NNODE * DDIM) + ng * 2 + dsel] = acc;
  }
}

extern "C" void kernel_launch(void* const* d_in, const int* in_sizes, int n_in,
                              void* d_out, int out_size, void* d_ws, size_t ws_size,
                              hipStream_t stream) {
  (void)in_sizes; (void)n_in; (void)out_size; (void)d_ws; (void)ws_size;
  const float* dq1 = (const float*)d_in[0];
  const float* dq2 = (const float*)d_in[1];
  const float* dp1 = (const float*)d_in[2];
  const float* dp2 = (const float*)d_in[3];
  const float* m   = (const float*)d_in[4];
  // d_in[5]=t, d_in[6]=dt : unused by the reference computation
  const float* len = (const float*)d_in[7];
  const float* kk  = (const float*)d_in[8];

  PSet pq, pp;
  {
    const float** f = (const float**)&pq;
    for (int i = 0; i < 18; ++i) f[i] = (const float*)d_in[9 + i];
    const float** g = (const float**)&pp;
    for (int i = 0; i < 18; ++i) g[i] = (const float*)d_in[27 + i];
  }

  const size_t smemBytes =
      (size_t)(NROWS * STR + EROWS * STR + WT_FLOATS) * sizeof(float);  // ~293 KB
  dim3 grid(B_TOT / BCHUNK, 2);
  gnet_fused_kernel<<<grid, 128, smemBytes, stream>>>(
      dq1, dq2, dp1, dp2, m, len, kk, pq, pp, (float*)d_out);
}